// STC_layer_89919435309240
// MI455X (gfx1250) — compile-verified
//
#include <hip/hip_runtime.h>

typedef __attribute__((ext_vector_type(2))) float v2f;
typedef __attribute__((ext_vector_type(8))) float v8f;

// ---------------------------------------------------------------------------
// Kernel 1: collapse the spectral filter into a single 32-coeff vector c.
//   a = U^T * avgweight          (32-vector, plain VALU, exact)
//   m = weight .* a              (32-vector)
//   c = U * m                    (32-vector, via V_WMMA_F32_16X16X4_F32)
// One wave32. EXEC is all ones through the WMMA region.
// WMMA mapping: D[16x16] = A[16x4] @ B[4x16] + C, chained over 8 K-chunks.
//   A layout (ISA 7.12.2, 32-bit A 16x4): lanes 0-15 hold M=lane, K={0,1} in
//   the two A VGPRs; lanes 16-31 hold M=lane-16, K={2,3}.
//   B is broadcast across N (all 16 columns identical) so every output column
//   holds c; lanes 0-15 carry K={0,1}, lanes 16-31 K={2,3}, mirroring A.
//   C/D layout: VGPR j -> M=j (lanes 0-15), M=j+8 (lanes 16-31).
// ---------------------------------------------------------------------------
__global__ __launch_bounds__(32) void stc_coeff_kernel(
    const float* __restrict__ U,    // [32,32] row-major (f,g)
    const float* __restrict__ w,    // [32]
    const float* __restrict__ av,   // [32]
    float* __restrict__ c_out) {    // [32]
  __shared__ float m_sh[32];
  const int lane = threadIdx.x;           // 0..31

  // a[g] = sum_f U[f,g] * av[f]   (lane == g)
  float a = 0.0f;
#pragma unroll
  for (int f = 0; f < 32; ++f) a += U[f * 32 + lane] * av[f];
  m_sh[lane] = w[lane] * a;
  __syncthreads();

  const int r15 = lane & 15;
  const int hi  = lane >> 4;              // 0: K={0,1}, 1: K={2,3} within chunk

#pragma unroll
  for (int h = 0; h < 2; ++h) {           // output rows f = 16h .. 16h+15
    v8f acc = {};
    const int row = h * 16 + r15;
#pragma unroll
    for (int kc = 0; kc < 8; ++kc) {      // K = 32 in chunks of 4
      const int kbase = kc * 4 + hi * 2;
      v2f A, B;
      A.x = U[row * 32 + kbase];
      A.y = U[row * 32 + kbase + 1];
      B.x = m_sh[kbase];                  // broadcast over all N columns
      B.y = m_sh[kbase + 1];
      acc = __builtin_amdgcn_wmma_f32_16x16x4_f32(
          /*neg_a=*/false, A, /*neg_b=*/false, B,
          /*c_mod=*/(short)0, acc, /*reuse_a=*/false, /*reuse_b=*/false);
    }
    if (r15 == 0) {                       // lanes 0 and 16 hold M=j / M=j+8
      const int mbase = h * 16 + hi * 8;
#pragma unroll
      for (int j = 0; j < 8; ++j) c_out[mbase + j] = acc[j];
    }
  }
}

// ---------------------------------------------------------------------------
// Kernel 2: out[b,d] = sum_{s=0..29} c[s+1] * feat[idx[b,s], d]
// Memory-bound weighted gather-sum: 307 MB of row reads (table is L2-resident:
// 102 MB < 192 MB global L2). float4 lanes -> 1 KB coalesced per row per
// 64-lane group; full unroll of the 30 rows keeps ~30 b128 loads in flight to
// hide random-access latency. blockDim = (64,4): 4 batch items / 256 threads.
// ---------------------------------------------------------------------------
__global__ __launch_bounds__(256) void stc_gather_kernel(
    const float4* __restrict__ feat,   // [N_table, 64] float4
    const int* __restrict__ idx,       // [B, 30] int32
    const float* __restrict__ c,       // [32] collapsed coefficients
    float4* __restrict__ out,          // [B, 64] float4
    int B) {
  const int b = blockIdx.x * blockDim.y + threadIdx.y;
  if (b >= B) return;
  const int t = threadIdx.x;           // 0..63 : float4 chunk of fdim=256
  const int* __restrict__ nb = idx + (size_t)b * 30;

  float4 acc = make_float4(0.f, 0.f, 0.f, 0.f);
#pragma unroll
  for (int s = 0; s < 30; ++s) {
    const int   row = nb[s];
    const float cs  = c[s + 1];        // slot 0 (center) and slot 31 are zero
    const float4 v  = feat[(size_t)row * 64 + t];
    acc.x = fmaf(cs, v.x, acc.x);
    acc.y = fmaf(cs, v.y, acc.y);
    acc.z = fmaf(cs, v.z, acc.z);
    acc.w = fmaf(cs, v.w, acc.w);
  }
  out[(size_t)b * 64 + t] = acc;
}

// ---------------------------------------------------------------------------
// Inputs (setup_inputs order):
//   0: feat_table  [100000*256] f32
//   1: neighbor_idx [B*30]      int32 (JAX x64 disabled -> int32)
//   2: weight      [32]         f32
//   3: avgweight   [32]         f32
//   4: U           [32*32]      f32
// Output: [B, 256] f32.  d_ws: c[32] scratch.
// ---------------------------------------------------------------------------
extern "C" void kernel_launch(void* const* d_in, const int* in_sizes, int n_in,
                              void* d_out, int out_size, void* d_ws, size_t ws_size,
                              hipStream_t stream) {
  const float* feat = (const float*)d_in[0];
  const int*   idx  = (const int*)d_in[1];
  const float* w    = (const float*)d_in[2];
  const float* av   = (const float*)d_in[3];
  const float* U    = (const float*)d_in[4];
  float* out = (float*)d_out;
  float* c   = (float*)d_ws;           // 32 floats of scratch

  const int S = in_sizes[2] - 2;       // 30 neighbors per node
  const int B = in_sizes[1] / S;       // 10000

  stc_coeff_kernel<<<1, 32, 0, stream>>>(U, w, av, c);

  dim3 blk(64, 4, 1);                  // 64 float4 chunks x 4 batch items
  dim3 grd((B + 3) / 4, 1, 1);
  stc_gather_kernel<<<grd, blk, 0, stream>>>(
      (const float4*)feat, idx, c, (float4*)out, B);
}